// Net_12962211299635
// MI455X (gfx1250) — compile-verified
//
#include <hip/hip_runtime.h>
#include <hip/hip_bf16.h>

// ---------------- problem constants (match reference) ----------------
#define N_NODES   50000
#define D_FEAT    512
#define H1_DIM    128
#define H2_DIM    64
#define N_CLS     40
#define E_MP      800000
#define E_LP      250000

typedef __attribute__((ext_vector_type(2))) float v2f;
typedef __attribute__((ext_vector_type(8))) float v8f;

// ---------------- small utility kernels ----------------
__global__ void fill_kernel(float* __restrict__ p, long long n, float v) {
  long long i = (long long)blockIdx.x * blockDim.x + threadIdx.x;
  if (i < n) p[i] = v;
}

__global__ void deg_accum_kernel(float* __restrict__ deg, const int* __restrict__ cols,
                                 const float* __restrict__ ew, int E) {
  long long i = (long long)blockIdx.x * blockDim.x + threadIdx.x;
  if (i < E) {
    float w = ew ? ew[i] : 1.0f;
    atomicAdd(&deg[cols[i]], w);
  }
}

__global__ void rsqrt_kernel(float* __restrict__ p, int n) {
  int i = blockIdx.x * blockDim.x + threadIdx.x;
  if (i < n) {
    float d = p[i];
    p[i] = (d > 0.0f) ? rsqrtf(d) : 0.0f;
  }
}

__global__ void bias_init_kernel(float* __restrict__ out, const float* __restrict__ b,
                                 long long n, int F) {
  long long i = (long long)blockIdx.x * blockDim.x + threadIdx.x;
  long long total = n * F;
  if (i < total) out[i] = b[i % F];
}

// ---------------- pack B[K,NC] into WMMA-fragment pair order, padded to NCP cols ----------
// pair p = (k2, col):  Bp[2p] = B[2*k2][col], Bp[2p+1] = B[2*k2+1][col]  (0 if col >= NC)
__global__ void pack_b_kernel(const float* __restrict__ B, float* __restrict__ Bp,
                              int K, int NC, int NCP) {
  int p = blockIdx.x * blockDim.x + threadIdx.x;
  int total = (K / 2) * NCP;
  if (p >= total) return;
  int k2 = p / NCP;
  int col = p % NCP;
  float x0 = 0.0f, x1 = 0.0f;
  if (col < NC) {
    x0 = B[(long long)(2 * k2) * NC + col];
    x1 = B[(long long)(2 * k2 + 1) * NC + col];
  }
  Bp[2 * p + 0] = x0;
  Bp[2 * p + 1] = x1;
}

// ---------------- WMMA fp32 GEMM: C[M,NCSTORE] = op(A[M,KDIM]) @ B[KDIM,NCSTORE] -----------
// B is pre-packed (pair-contiguous, NCP = NT*16 columns). One wave computes a 16-row stripe
// across all NT 16-wide column tiles, sharing one A fragment per k-step.
// A frag (16x4 f32): lanes 0-15 -> M=0..15, K={0,1}; lanes 16-31 -> K={2,3}.
// B frag (4x16 f32): lanes 0-15 -> N=0..15, K={0,1}; lanes 16-31 -> K={2,3}.
// C/D (16x16 f32):   VGPR r, lanes 0-15 -> M=r; lanes 16-31 -> M=r+8.
template <int KDIM, int NT, int NCSTORE, bool RELU_A>
__global__ void wmma_gemm_kernel(const float* __restrict__ A, const float* __restrict__ Bp,
                                 float* __restrict__ C, int M) {
  constexpr int NCP = NT * 16;
  int wave = (int)(((long long)blockIdx.x * blockDim.x + threadIdx.x) >> 5);
  int lane = threadIdx.x & 31;
  int ntiles = M >> 4;              // M multiple of 16 (50000 = 3125*16)
  if (wave >= ntiles) return;       // wave-uniform exit: EXEC all-1s inside WMMA

  int row0 = wave << 4;
  int arow = row0 + (lane & 15);
  int khb  = lane >> 4;             // 0 or 1  (khalf = 2*khb)
  int bcol = lane & 15;

  v8f acc[NT] = {};

  const float2* __restrict__ Arow2 = (const float2*)(A + (long long)arow * KDIM);
  const float2* __restrict__ Bp2   = (const float2*)Bp;

#pragma unroll 2
  for (int k2 = 0; k2 < KDIM / 2; k2 += 2) {   // k2 = k/2, steps of 2 pairs = K step 4
    // clause-load NT B fragments (constant offsets from one base) + the A fragment
    const float2* __restrict__ Bbase = Bp2 + (long long)(k2 + khb) * NCP + bcol;
    float2 bf[NT];
#pragma unroll
    for (int t = 0; t < NT; ++t) bf[t] = Bbase[t * 16];

    float2 af = Arow2[k2 + khb];
    if (RELU_A) { af.x = fmaxf(af.x, 0.0f); af.y = fmaxf(af.y, 0.0f); }
    v2f a; a.x = af.x; a.y = af.y;

#pragma unroll
    for (int t = 0; t < NT; ++t) {
      v2f b; b.x = bf[t].x; b.y = bf[t].y;
      acc[t] = __builtin_amdgcn_wmma_f32_16x16x4_f32(
          /*neg_a=*/false, a, /*neg_b=*/false, b,
          /*c_mod=*/(short)0, acc[t], /*reuse_a=*/false, /*reuse_b=*/false);
    }
  }

  int crow = row0 + (khb << 3);     // +0 or +8
  int ccol = lane & 15;
#pragma unroll
  for (int t = 0; t < NT; ++t) {
    int col = (t << 4) + ccol;
    if (col < NCSTORE) {            // resolves statically for full-width layers
#pragma unroll
      for (int r = 0; r < 8; ++r) {
        C[(long long)(crow + r) * NCSTORE + col] = acc[t][r];
      }
    }
  }
}

// ---------------- GCN edge aggregation: out[col] += dinv[row]*w*dinv[col] * hx[row] ----------
// Work item = (edge-or-selfloop, 4-feature chunk). All buffers are L2-resident (<192MB).
__global__ void gcn_agg_kernel(const float* __restrict__ hx, float* __restrict__ out,
                               const int* __restrict__ rows, const int* __restrict__ cols,
                               const float* __restrict__ ew, const float* __restrict__ dinv,
                               int E, int N, int F) {
  int chunks = F >> 2;
  long long tid = (long long)blockIdx.x * blockDim.x + threadIdx.x;
  long long total = (long long)(E + N) * chunks;
  if (tid >= total) return;
  int e = (int)(tid / chunks);
  int c = (int)(tid % chunks) << 2;
  int r, cl; float w;
  if (e < E) { r = rows[e]; cl = cols[e]; w = ew ? ew[e] : 1.0f; }
  else       { r = e - E;   cl = r;       w = 1.0f; }
  float norm = dinv[r] * w * dinv[cl];
  const float4 h4 = *(const float4*)(hx + (long long)r * F + c);
  float* o = out + (long long)cl * F + c;
  atomicAdd(o + 0, h4.x * norm);
  atomicAdd(o + 1, h4.y * norm);
  atomicAdd(o + 2, h4.z * norm);
  atomicAdd(o + 3, h4.w * norm);
}

// ---------------- per-row log_softmax over F classes ----------------
__global__ void log_softmax_kernel(const float* __restrict__ in, float* __restrict__ out,
                                   int n, int F) {
  int row = blockIdx.x * blockDim.x + threadIdx.x;
  if (row >= n) return;
  const float* p = in + (long long)row * F;
  float m = -INFINITY;
  for (int c = 0; c < F; ++c) m = fmaxf(m, p[c]);
  float s = 0.0f;
  for (int c = 0; c < F; ++c) s += __expf(p[c] - m);
  float ls = __logf(s) + m;
  float* o = out + (long long)row * F;
  for (int c = 0; c < F; ++c) o[c] = p[c] - ls;
}

// ---------------- link-prediction edge dot products ----------------
__global__ void edge_dot_kernel(const float* __restrict__ h, const int* __restrict__ pos,
                                const int* __restrict__ neg, float* __restrict__ res) {
  int e = blockIdx.x * blockDim.x + threadIdx.x;
  if (e >= 2 * E_LP) return;
  const int* src = (e < E_LP) ? pos : neg;
  int eo = (e < E_LP) ? e : (e - E_LP);
  int j = src[eo];         // total[0][e]
  int i = src[E_LP + eo];  // total[1][e]
  const float4* hj = (const float4*)(h + (long long)j * H2_DIM);
  const float4* hi = (const float4*)(h + (long long)i * H2_DIM);
  float s = 0.0f;
#pragma unroll
  for (int k = 0; k < H2_DIM / 4; ++k) {
    float4 a = hi[k], b = hj[k];
    s += a.x * b.x + a.y * b.y + a.z * b.z + a.w * b.w;
  }
  res[e] = s;
}

// ---------------- host-side launcher ----------------
static inline int blocks_for(long long total, int bs) {
  return (int)((total + bs - 1) / bs);
}

extern "C" void kernel_launch(void* const* d_in, const int* in_sizes, int n_in,
                              void* d_out, int out_size, void* d_ws, size_t ws_size,
                              hipStream_t stream) {
  const float* x      = (const float*)d_in[0];
  const int*   tpe    = (const int*)  d_in[1];   // [2, E_MP]
  const int*   ei     = (const int*)  d_in[2];   // [2, E_MP]
  const float* ew     = (const float*)d_in[3];   // [E_MP]
  const int*   pos    = (const int*)  d_in[4];   // [2, E_LP]
  const int*   neg    = (const int*)  d_in[5];   // [2, E_LP]
  const float* W1     = (const float*)d_in[6];
  const float* b1     = (const float*)d_in[7];
  const float* W2     = (const float*)d_in[8];
  const float* b2     = (const float*)d_in[9];
  const float* W_attr = (const float*)d_in[10];
  const float* b_attr = (const float*)d_in[11];
  const float* W_att  = (const float*)d_in[12];
  const float* b_att  = (const float*)d_in[13];

  float* out = (float*)d_out;
  float* ws  = (float*)d_ws;

  // workspace layout (floats); peak footprint ~78 MB -> fits in 192 MB L2
  const long long OFF_DINV_T = 0;                          // 50000
  const long long OFF_DINV_W = 50000;                      // 50000
  const long long OFF_HX1    = 100000;                     // 50000*128
  const long long OFF_H1     = OFF_HX1 + 6400000;          // 50000*128
  const long long OFF_HX2    = OFF_H1  + 6400000;          // 50000*64
  const long long OFF_H2     = OFF_HX2 + 3200000;          // 50000*64
  const long long OFF_W1P    = OFF_H2  + 3200000;          // 256*128*2 = 65536
  const long long OFF_W2P    = OFF_W1P + 65536;            // 64*64*2   = 8192
  const long long OFF_WAP    = OFF_W2P + 8192;             // 32*48*2   = 3072
  const long long OFF_WBP    = OFF_WAP + 3072;             // 32*48*2   = 3072
  // region reuse after each stage retires:
  const long long OFF_HXA    = OFF_HX1;                    // 50000*40 (hx1 free)
  const long long OFF_HXB    = OFF_HX1 + 2000000;          // 50000*40
  const long long OFF_ATTR   = OFF_H1;                     // 50000*40 (h1 free)
  const long long OFF_ATT    = OFF_H1 + 2000000;           // 50000*40

  float* dinv_t = ws + OFF_DINV_T;
  float* dinv_w = ws + OFF_DINV_W;
  float* hx1    = ws + OFF_HX1;
  float* h1     = ws + OFF_H1;
  float* hx2    = ws + OFF_HX2;
  float* h2     = ws + OFF_H2;
  float* w1p    = ws + OFF_W1P;
  float* w2p    = ws + OFF_W2P;
  float* wap    = ws + OFF_WAP;
  float* wbp    = ws + OFF_WBP;
  float* hxa    = ws + OFF_HXA;
  float* hxb    = ws + OFF_HXB;
  float* attr_a = ws + OFF_ATTR;
  float* att_a  = ws + OFF_ATT;

  const int BS = 256;

  // ---- pack weights into WMMA fragment order (heads padded 40 -> 48 cols) ----
  pack_b_kernel<<<blocks_for((D_FEAT / 2) * H1_DIM, BS), BS, 0, stream>>>(W1, w1p, D_FEAT, H1_DIM, H1_DIM);
  pack_b_kernel<<<blocks_for((H1_DIM / 2) * H2_DIM, BS), BS, 0, stream>>>(W2, w2p, H1_DIM, H2_DIM, H2_DIM);
  pack_b_kernel<<<blocks_for((H2_DIM / 2) * 48, BS), BS, 0, stream>>>(W_attr, wap, H2_DIM, N_CLS, 48);
  pack_b_kernel<<<blocks_for((H2_DIM / 2) * 48, BS), BS, 0, stream>>>(W_att,  wbp, H2_DIM, N_CLS, 48);

  // ---- symmetric norm factors: deg init 1 (self loop) + sum over edges, then rsqrt ----
  fill_kernel<<<blocks_for(N_NODES, BS), BS, 0, stream>>>(dinv_t, N_NODES, 1.0f);
  deg_accum_kernel<<<blocks_for(E_MP, BS), BS, 0, stream>>>(dinv_t, tpe + E_MP, nullptr, E_MP);
  rsqrt_kernel<<<blocks_for(N_NODES, BS), BS, 0, stream>>>(dinv_t, N_NODES);

  fill_kernel<<<blocks_for(N_NODES, BS), BS, 0, stream>>>(dinv_w, N_NODES, 1.0f);
  deg_accum_kernel<<<blocks_for(E_MP, BS), BS, 0, stream>>>(dinv_w, ei + E_MP, ew, E_MP);
  rsqrt_kernel<<<blocks_for(N_NODES, BS), BS, 0, stream>>>(dinv_w, N_NODES);

  const int ROW_TILES = N_NODES / 16;                    // 3125
  const int GEMM_BS = 128;                               // 4 waves per block
  const int gemm_blocks = blocks_for((long long)ROW_TILES * 32, GEMM_BS);

  // ---- layer 1: hx1 = x @ W1 ; h1 = b1 + norm-agg(hx1) ----
  wmma_gemm_kernel<D_FEAT, 8, H1_DIM, false><<<gemm_blocks, GEMM_BS, 0, stream>>>(x, w1p, hx1, N_NODES);
  bias_init_kernel<<<blocks_for((long long)N_NODES * H1_DIM, BS), BS, 0, stream>>>(h1, b1, N_NODES, H1_DIM);
  gcn_agg_kernel<<<blocks_for((long long)(E_MP + N_NODES) * (H1_DIM / 4), BS), BS, 0, stream>>>(
      hx1, h1, tpe, tpe + E_MP, nullptr, dinv_t, E_MP, N_NODES, H1_DIM);

  // ---- layer 2: hx2 = relu(h1) @ W2 ; h2 = b2 + norm-agg(hx2) ----
  wmma_gemm_kernel<H1_DIM, 4, H2_DIM, true><<<gemm_blocks, GEMM_BS, 0, stream>>>(h1, w2p, hx2, N_NODES);
  bias_init_kernel<<<blocks_for((long long)N_NODES * H2_DIM, BS), BS, 0, stream>>>(h2, b2, N_NODES, H2_DIM);
  gcn_agg_kernel<<<blocks_for((long long)(E_MP + N_NODES) * (H2_DIM / 4), BS), BS, 0, stream>>>(
      hx2, h2, tpe, tpe + E_MP, nullptr, dinv_t, E_MP, N_NODES, H2_DIM);

  // ---- heads: attr = gcn(h2, W_attr) over weighted graph; att likewise ----
  wmma_gemm_kernel<H2_DIM, 3, N_CLS, false><<<gemm_blocks, GEMM_BS, 0, stream>>>(h2, wap, hxa, N_NODES);
  wmma_gemm_kernel<H2_DIM, 3, N_CLS, false><<<gemm_blocks, GEMM_BS, 0, stream>>>(h2, wbp, hxb, N_NODES);

  bias_init_kernel<<<blocks_for((long long)N_NODES * N_CLS, BS), BS, 0, stream>>>(attr_a, b_attr, N_NODES, N_CLS);
  gcn_agg_kernel<<<blocks_for((long long)(E_MP + N_NODES) * (N_CLS / 4), BS), BS, 0, stream>>>(
      hxa, attr_a, ei, ei + E_MP, ew, dinv_w, E_MP, N_NODES, N_CLS);

  bias_init_kernel<<<blocks_for((long long)N_NODES * N_CLS, BS), BS, 0, stream>>>(att_a, b_att, N_NODES, N_CLS);
  gcn_agg_kernel<<<blocks_for((long long)(E_MP + N_NODES) * (N_CLS / 4), BS), BS, 0, stream>>>(
      hxb, att_a, ei, ei + E_MP, ew, dinv_w, E_MP, N_NODES, N_CLS);

  // ---- outputs: res [500000] | log_softmax(attr) [2e6] | log_softmax(att) [2e6] ----
  edge_dot_kernel<<<blocks_for(2 * E_LP, BS), BS, 0, stream>>>(h2, pos, neg, out);
  log_softmax_kernel<<<blocks_for(N_NODES, BS), BS, 0, stream>>>(attr_a, out + 2 * E_LP, N_NODES, N_CLS);
  log_softmax_kernel<<<blocks_for(N_NODES, BS), BS, 0, stream>>>(att_a,  out + 2 * E_LP + (long long)N_NODES * N_CLS, N_NODES, N_CLS);
}